// CapsuleLayer_19696720019935
// MI455X (gfx1250) — compile-verified
//
#include <hip/hip_runtime.h>

// CapsuleLayer dynamic routing for MI455X (gfx1250).
// - u_hat never materialized (268MB); recomputed per routing round with
//   V_WMMA_F32_16X16X4_F32 (exact f32). W (16.8MB) + inputs (8.4MB) are
//   L2-resident, so passes are L2-bound, not HBM-bound (23.3 TB/s never the
//   limiter; FLOPs ~1 GF/pass are trivial for the matrix engines).
// - W tiles are DMA'd into LDS with the Tensor Data Mover (tensor_load_to_lds,
//   TENSORcnt double-buffering): all 8 waves of a block share one copy instead
//   of 8x redundant vector loads.
// - Routing logits use b_r = (sum of previous v) . u_hat, so only a 128KB
//   cumulative Vcum is kept. Workspace use: 256KB.

typedef float v2f __attribute__((ext_vector_type(2)));
typedef float v4f __attribute__((ext_vector_type(4)));
typedef float v8f __attribute__((ext_vector_type(8)));
typedef unsigned int u32x4 __attribute__((ext_vector_type(4)));
typedef int i32x4 __attribute__((ext_vector_type(4)));
typedef int i32x8 __attribute__((ext_vector_type(8)));

namespace {
constexpr int kB = 128;   // batch
constexpr int kI = 2048;  // input capsules
constexpr int kF = 8;     // input capsule dim (GEMM K)
constexpr int kJ = 16;    // output capsules
constexpr int kD = 16;    // output capsule dim (GEMM M)
constexpr int kIChunk = 4;          // i's per TDM tile (32KB W tile)
constexpr int kChunks = 4;          // chunks per block (double-buffered)
constexpr int kBlocks = kI / (kIChunk * kChunks);  // 128
constexpr int kNS = kB * kJ * kD;   // 32768 elements in s / Vcum
constexpr int kWTile = kJ * kIChunk * kD * kF;     // 8192 floats = 32KB
}

__global__ void caps_zero_kernel(float* __restrict__ p, int n) {
  int t = blockIdx.x * blockDim.x + threadIdx.x;
  if (t < n) p[t] = 0.0f;
}

// Issue one TDM load of W[j=0..15, i0..i0+3, d, f] into LDS at lds_off.
// D# tile: X = 128 contiguous (d*8+f) elements, Y = i (stride 128 elems),
// Z = j (stride I*128 = 262144 elems). LDS receives [j][il][d][f] contiguous.
__device__ __forceinline__ void tdm_load_w_tile(const float* Wg, int i0,
                                                unsigned lds_off) {
  unsigned long long ga =
      (unsigned long long)(uintptr_t)Wg +
      (unsigned long long)i0 * (unsigned long long)(kD * kF * 4);
  u32x4 g0;
  g0.x = 1u;                                   // count=1 (valid descriptor)
  g0.y = lds_off;                              // lds_addr (bytes)
  g0.z = (unsigned)(ga & 0xFFFFFFFFull);       // global_addr[31:0]
  g0.w = (unsigned)((ga >> 32) & 0x1FFFFFFull) // global_addr[56:32]
         | (2u << 30);                         // type=2 ("image")
  i32x8 g1;
  g1[0] = (int)(2u << 16);                     // data_size=2 -> 4 bytes
  g1[1] = (int)(128u << 16);                   // tensor_dim0[15:0]=128
  g1[2] = (int)(2048u << 16);                  // tensor_dim1[15:0]=2048
  g1[3] = (int)(128u << 16);                   // tile_dim0=128
  g1[4] = (int)((unsigned)kIChunk | (16u << 16)); // tile_dim1=4, tile_dim2=16
  g1[5] = 128;                                 // tensor_dim0_stride=128
  g1[6] = 0;                                   // (262144 & 0xFFFF)<<16 == 0
  g1[7] = (int)(262144u >> 16);                // tensor_dim1_stride[47:16]=4
  i32x4 g2;
  g2[0] = 16;  // tensor_dim2 = 16 (j)
  g2[1] = 0;   // tensor_dim3 unused
  g2[2] = 0;
  g2[3] = 0;   // tile_dim3 = 0
  i32x4 g3;
  g3[0] = 0; g3[1] = 0; g3[2] = 0; g3[3] = 0;
  i32x8 g4;    // extra group required by the 6-arg toolchain builtin; zeros
  g4[0] = 0; g4[1] = 0; g4[2] = 0; g4[3] = 0;
  g4[4] = 0; g4[5] = 0; g4[6] = 0; g4[7] = 0;
  __builtin_amdgcn_tensor_load_to_lds(g0, g1, g2, g3, g4, 0);
}

// One wave = one 16-wide b-tile; 8 waves/block cover all 128 b for an i-range.
// WMMA per (j,i): A = W[j,i,d,f] (16x8 as two 16x4, from LDS), B =
// inputs[b,i,f]^T (8x16). C/D layout: lane l -> (N=b_local=l%16,
// M=d=vgpr_k+8*(l/16)).
template <bool UNIFORM>
__global__ __launch_bounds__(256, 1)
void caps_accum_kernel(const float* __restrict__ inputs,
                       const float* __restrict__ W,
                       const float* __restrict__ Vcum,
                       float* __restrict__ s_glob) {
  __shared__ float lds_w[2][kWTile];  // 2 x 32KB double buffer

  const int lane = threadIdx.x & 31;
  const int wave = threadIdx.x >> 5;
  const int bl   = lane & 15;  // b within tile / N index
  const int hi   = lane >> 4;  // lane half (selects K pair / upper M half)
  const int b    = wave * 16 + bl;
  const int ibase = blockIdx.x * (kIChunk * kChunks);

  v8f s_acc[kJ];
#pragma unroll
  for (int j = 0; j < kJ; ++j)
    s_acc[j] = v8f{0.f, 0.f, 0.f, 0.f, 0.f, 0.f, 0.f, 0.f};

  const unsigned lds0 = (unsigned)(uintptr_t)&lds_w[0][0];
  const unsigned lds1 = (unsigned)(uintptr_t)&lds_w[1][0];

  // Prologue: DMA chunk 0 into buffer 0 (wave 0 owns the TDM + TENSORcnt).
  if (wave == 0) tdm_load_w_tile(W, ibase, lds0);

  for (int c = 0; c < kChunks; ++c) {
    const int buf = c & 1;
    if (wave == 0) {
      if (c + 1 < kChunks) {
        tdm_load_w_tile(W, ibase + (c + 1) * kIChunk, (c & 1) ? lds0 : lds1);
        __builtin_amdgcn_s_wait_tensorcnt((short)1);  // chunk c done (in-order)
      } else {
        __builtin_amdgcn_s_wait_tensorcnt((short)0);
      }
    }
    __syncthreads();  // W tile for chunk c visible to all waves

    for (int il = 0; il < kIChunk; ++il) {
      const int i = ibase + c * kIChunk + il;

      // B fragments (shared across all j): K rows f = 2*hi+{0,1} (+4 chunk1)
      const float* ib = inputs + ((size_t)b * kI + i) * kF + 2 * hi;
      const v2f bf0 = *(const v2f*)ib;
      const v2f bf1 = *(const v2f*)(ib + 4);

      float cj[kJ];
      if constexpr (!UNIFORM) {
        // ---- Pass A: logits lg[j] = Vcum[b,j,:] . u_hat[b,j,i,:] ----
        float lg[kJ];
#pragma unroll
        for (int j = 0; j < kJ; ++j) {
          const float* wl =
              &lds_w[buf][((j * kIChunk + il) * kD + bl) * kF + 2 * hi];
          const v2f a0 = *(const v2f*)wl;
          const v2f a1 = *(const v2f*)(wl + 4);
          v8f u = v8f{0.f, 0.f, 0.f, 0.f, 0.f, 0.f, 0.f, 0.f};
          u = __builtin_amdgcn_wmma_f32_16x16x4_f32(false, a0, false, bf0,
                                                    (short)0, u, false, false);
          u = __builtin_amdgcn_wmma_f32_16x16x4_f32(false, a1, false, bf1,
                                                    (short)0, u, false, false);
          // Vcum fragment in matching C layout: d = k + 8*hi (contiguous)
          const float* vc = Vcum + ((size_t)b * kJ + j) * kD + 8 * hi;
          const v4f vca = *(const v4f*)vc;
          const v4f vcb = *(const v4f*)(vc + 4);
          float p = 0.f;
#pragma unroll
          for (int k = 0; k < 4; ++k) p = fmaf(u[k], vca[k], p);
#pragma unroll
          for (int k = 0; k < 4; ++k) p = fmaf(u[k + 4], vcb[k], p);
          p += __shfl_xor(p, 16, 32);  // combine the two d-halves (wave32)
          lg[j] = p;
        }
        // ---- softmax over j (per lane; both halves identical) ----
        float m = lg[0];
#pragma unroll
        for (int j = 1; j < kJ; ++j) m = fmaxf(m, lg[j]);
        float den = 0.f;
#pragma unroll
        for (int j = 0; j < kJ; ++j) {
          cj[j] = __expf(lg[j] - m);
          den += cj[j];
        }
        const float inv = 1.0f / den;
#pragma unroll
        for (int j = 0; j < kJ; ++j) cj[j] *= inv;
      }

      // ---- Pass B: s[b,j,:] += c[b,j,i] * u_hat[b,j,i,:] ----
#pragma unroll
      for (int j = 0; j < kJ; ++j) {
        const float* wl =
            &lds_w[buf][((j * kIChunk + il) * kD + bl) * kF + 2 * hi];
        const v2f a0 = *(const v2f*)wl;
        const v2f a1 = *(const v2f*)(wl + 4);
        if constexpr (UNIFORM) {
          // c uniform (1/16, folded into squash): chain accumulator over i
          s_acc[j] = __builtin_amdgcn_wmma_f32_16x16x4_f32(
              false, a0, false, bf0, (short)0, s_acc[j], false, false);
          s_acc[j] = __builtin_amdgcn_wmma_f32_16x16x4_f32(
              false, a1, false, bf1, (short)0, s_acc[j], false, false);
        } else {
          v8f u = v8f{0.f, 0.f, 0.f, 0.f, 0.f, 0.f, 0.f, 0.f};
          u = __builtin_amdgcn_wmma_f32_16x16x4_f32(false, a0, false, bf0,
                                                    (short)0, u, false, false);
          u = __builtin_amdgcn_wmma_f32_16x16x4_f32(false, a1, false, bf1,
                                                    (short)0, u, false, false);
          const float cw = cj[j];
#pragma unroll
          for (int k = 0; k < 8; ++k) s_acc[j][k] = fmaf(cw, u[k], s_acc[j][k]);
        }
      }
    }
    __syncthreads();  // all waves done reading buf before it is re-filled
  }

  // Flush block-local partial s with fp32 global atomics.
#pragma unroll
  for (int j = 0; j < kJ; ++j) {
#pragma unroll
    for (int k = 0; k < 8; ++k) {
      float* dst = s_glob + ((size_t)b * kJ + j) * kD + (8 * hi + k);
      __hip_atomic_fetch_add(dst, s_acc[j][k], __ATOMIC_RELAXED,
                             __HIP_MEMORY_SCOPE_AGENT);
    }
  }
}

// squash(s) = sq/(1+sq)/sqrt(sq+eps) * s ; writes v and updates Vcum.
// pre = 1/16 for round 0 (uniform routing weights folded here).
__global__ __launch_bounds__(256)
void caps_squash_kernel(const float* __restrict__ s, float* __restrict__ Vcum,
                        float* __restrict__ vout, float pre, int first) {
  int bj = blockIdx.x * blockDim.x + threadIdx.x;
  if (bj >= kB * kJ) return;
  const float* sp = s + (size_t)bj * kD;
  float sv[kD];
  float sq = 0.f;
#pragma unroll
  for (int d = 0; d < kD; ++d) {
    sv[d] = sp[d] * pre;
    sq = fmaf(sv[d], sv[d], sq);
  }
  const float scale = (sq / (1.0f + sq)) * rsqrtf(sq + 1e-7f);
  float* vc = Vcum + (size_t)bj * kD;
  float* vo = vout + (size_t)bj * kD;
#pragma unroll
  for (int d = 0; d < kD; ++d) {
    const float v = scale * sv[d];
    vo[d] = v;
    vc[d] = first ? v : (vc[d] + v);
  }
}

extern "C" void kernel_launch(void* const* d_in, const int* in_sizes, int n_in,
                              void* d_out, int out_size, void* d_ws,
                              size_t ws_size, hipStream_t stream) {
  const float* inputs = (const float*)d_in[0];  // [128, 2048, 8]
  const float* W      = (const float*)d_in[1];  // [16, 2048, 16, 8]
  float* out = (float*)d_out;                   // [128, 16, 16]

  float* s    = (float*)d_ws;  // 32768 f32
  float* Vcum = s + kNS;       // 32768 f32

  const dim3 gz((kNS + 255) / 256), bz(256);
  const dim3 ga(kBlocks), ba(256);
  const dim3 gq((kB * kJ + 255) / 256), bq(256);

  // Round 0: c uniform -> s = (1/16) * sum_i u_hat (scale folded into squash)
  caps_zero_kernel<<<gz, bz, 0, stream>>>(s, kNS);
  caps_accum_kernel<true><<<ga, ba, 0, stream>>>(inputs, W, Vcum, s);
  caps_squash_kernel<<<gq, bq, 0, stream>>>(s, Vcum, out, 1.0f / 16.0f, 1);

  // Round 1: logits = v0 . u_hat
  caps_zero_kernel<<<gz, bz, 0, stream>>>(s, kNS);
  caps_accum_kernel<false><<<ga, ba, 0, stream>>>(inputs, W, Vcum, s);
  caps_squash_kernel<<<gq, bq, 0, stream>>>(s, Vcum, out, 1.0f, 0);

  // Round 2: logits = (v0+v1) . u_hat ; final v written to d_out
  caps_zero_kernel<<<gz, bz, 0, stream>>>(s, kNS);
  caps_accum_kernel<false><<<ga, ba, 0, stream>>>(inputs, W, Vcum, s);
  caps_squash_kernel<<<gq, bq, 0, stream>>>(s, Vcum, out, 1.0f, 0);
}